// ElevationLoss_87754771792334
// MI455X (gfx1250) — compile-verified
//
#include <hip/hip_runtime.h>
#include <hip/hip_bf16.h>

// ---------------------------------------------------------------------------
// ElevationLoss for MI455X (gfx1250, wave32).
// Memory-bound single-pass reduction: 16 MiB of f32 inputs, scalar f32 out.
// LDS-tiled 3x3 neighborhood reuse + WMMA (V_WMMA_F32_16X16X4_F32) wave
// reduction + f64 atomics for a deterministic-enough, precise final sum.
// ---------------------------------------------------------------------------

typedef float v2f __attribute__((ext_vector_type(2)));
typedef float v8f __attribute__((ext_vector_type(8)));

#define TS 16          // tile side; block = 256 threads = 8 waves
#define HALO (TS + 2)  // 18x18 halo of gt / heights

// Full-wave (32 lane) sum using V_WMMA_F32_16X16X4_F32.
// A-layout (ISA 7.12.2, 32-bit A 16x4): lane L holds A[M=L%16][K=2*(L/16)]
// in VGPR0 and A[M][K=2*(L/16)+1] in VGPR1. We park the lane value in VGPR0
// and 0 in VGPR1, use an all-ones B (layout independent), so
//   D[m,n] = sum_k A[m,k] = v_m + v_{m+16}  for every n.
// C/D layout: VGPR r holds M=r (lanes 0-15) / M=r+8 (lanes 16-31), so the
// per-lane sum of the 8 D VGPRs gives the half sum; one xor-16 shuffle
// finishes the reduction. Requires EXEC all-ones (call from all 256 threads).
__device__ __forceinline__ float wave_sum_wmma(float v) {
    v2f a;
    a[0] = v;
    a[1] = 0.0f;
    v2f b;
    b[0] = 1.0f;
    b[1] = 1.0f;
    v8f c = {0.0f, 0.0f, 0.0f, 0.0f, 0.0f, 0.0f, 0.0f, 0.0f};
    v8f d = __builtin_amdgcn_wmma_f32_16x16x4_f32(
        /*neg_a=*/false, a, /*neg_b=*/false, b,
        /*c_mod=*/(short)0, c, /*reuse_a=*/false, /*reuse_b=*/false);
    float s = d[0] + d[1] + d[2] + d[3] + d[4] + d[5] + d[6] + d[7];
    s += __shfl_xor(s, 16, 32);   // combine M=0..7 half with M=8..15 half
    return s;                      // full 32-lane sum, in every lane
}

__global__ void elev_ws_init(double* acc) {
    acc[0] = 0.0;
    acc[1] = 0.0;
}

__global__ __launch_bounds__(256) void elev_loss_kernel(
    const float* __restrict__ pred,   // (B,2,H,W)
    const float* __restrict__ hts,    // (B,1,H,W)
    const float* __restrict__ gt,     // (B,1,H,W) values in {0,1,255}
    double* __restrict__ acc,         // acc[0]=loss sum, acc[1]=valid count
    int H, int W) {
    __shared__ float sg[HALO][HALO + 1];  // gt halo (+1 pad vs bank conflicts)
    __shared__ float sh[HALO][HALO + 1];  // heights halo
    __shared__ float wpart[2][8];         // per-wave partials (loss, cnt)

    const int b  = blockIdx.z;
    const int bx = blockIdx.x * TS;
    const int by = blockIdx.y * TS;
    const int tid = threadIdx.x;

    const size_t hw = (size_t)H * (size_t)W;
    const float* gtb = gt  + (size_t)b * hw;
    const float* hb  = hts + (size_t)b * hw;
    const float* pb  = pred + (size_t)b * 2 * hw;

    // Cooperative halo load: 18x18 = 324 elements, 256 threads, 2 passes.
    // Out-of-image cells are the zero padding of the reference unfold:
    // gt=0 (valid, dry) and height=0.
    for (int i = tid; i < HALO * HALO; i += 256) {
        int r = i / HALO, q = i % HALO;
        int gy = by - 1 + r;
        int gx = bx - 1 + q;
        bool ib = (gy >= 0) & (gy < H) & (gx >= 0) & (gx < W);
        size_t idx = (size_t)(ib ? gy : 0) * W + (ib ? gx : 0);
        sg[r][q] = ib ? gtb[idx] : 0.0f;
        sh[r][q] = ib ? hb[idx]  : 0.0f;
    }
    __syncthreads();

    const int lx = tid % TS;
    const int ly = tid / TS;
    const int y = by + ly;
    const int x = bx + lx;
    const bool live = (y < H) & (x < W);   // always true for 512x512 grid
    const size_t pix = (size_t)(live ? y : 0) * W + (live ? x : 0);

    // unified prediction: argmax tie-break matches jnp.argmax (first index)
    float p0 = pb[pix];
    float p1 = pb[hw + pix];
    float u = (p0 >= p1) ? p0 : -p1;
    float hc = sh[ly + 1][lx + 1];

    float loss = 0.0f;
    float cnt = 0.0f;
#pragma unroll
    for (int di = 0; di < 3; ++di) {
#pragma unroll
        for (int dj = 0; dj < 3; ++dj) {
            float g  = sg[ly + di][lx + dj];
            float hn = sh[ly + di][lx + dj];
            float ig = (g != 255.0f) ? 1.0f : 0.0f;             // ignore_mask
            float gr = (g == 1.0f) ? 1.0f
                                   : ((g == 0.0f) ? -1.0f : 0.0f);  // remapped
            float delta = hc - hn;
            float fpe = ((g == 1.0f) & (delta > 0.0f)) ? 0.0f : 1.0f;
            float dne = ((g == 0.0f) & (delta < 0.0f)) ? 0.0f : 1.0f;
            float score = 1.0f - gr * u;
            loss += ig * fpe * dne * score;
            cnt  += ig;
        }
    }
    if (!live) { loss = 0.0f; cnt = 0.0f; }

    // Wave reduction via WMMA (EXEC all-ones here: no divergence at this
    // point, every thread participates).
    float wl = wave_sum_wmma(loss);
    float wc = wave_sum_wmma(cnt);

    const int wave = tid >> 5;
    const int lane = tid & 31;
    if (lane == 0) {
        wpart[0][wave] = wl;
        wpart[1][wave] = wc;
    }
    __syncthreads();

    if (tid == 0) {
        float bl = 0.0f, bc = 0.0f;
#pragma unroll
        for (int w = 0; w < 8; ++w) {
            bl += wpart[0][w];
            bc += wpart[1][w];
        }
        // f64 accumulation: exact for the ~9.4M valid count (beyond f32's
        // exact-integer range in aggregate) and order-insensitive enough for
        // the loss sum. ~4096 atomics total.
        atomicAdd(&acc[0], (double)bl);
        atomicAdd(&acc[1], (double)bc);
    }
}

__global__ void elev_finalize(const double* __restrict__ acc,
                              float* __restrict__ out) {
    double n = acc[1];
    if (n < 1.0) n = 1.0;
    out[0] = (float)(acc[0] / n);
}

extern "C" void kernel_launch(void* const* d_in, const int* in_sizes, int n_in,
                              void* d_out, int out_size, void* d_ws, size_t ws_size,
                              hipStream_t stream) {
    const float* pred = (const float*)d_in[0];  // (B,2,H,W)
    const float* hts  = (const float*)d_in[1];  // (B,1,H,W)
    const float* gt   = (const float*)d_in[2];  // (B,1,H,W)
    float* out = (float*)d_out;
    double* acc = (double*)d_ws;

    const int H = 512, W = 512;
    const int B = in_sizes[1] / (H * W);  // heights element count / HW

    elev_ws_init<<<1, 1, 0, stream>>>(acc);

    dim3 grid((W + TS - 1) / TS, (H + TS - 1) / TS, B);  // 32 x 32 x 4
    dim3 block(256);
    elev_loss_kernel<<<grid, block, 0, stream>>>(pred, hts, gt, acc, H, W);

    elev_finalize<<<1, 1, 0, stream>>>(acc, out);
}